// KLDivergenceLoss_317827580600
// MI455X (gfx1250) — compile-verified
//
#include <hip/hip_runtime.h>
#include <math.h>

#ifndef __has_builtin
#define __has_builtin(x) 0
#endif

#define TILE 256
#define VEC 21

#if __has_builtin(__builtin_amdgcn_rsqf)
#define RSQF(x) __builtin_amdgcn_rsqf(x)
#else
#define RSQF(x) (1.0f / sqrtf(x))
#endif

// Vector/pointer types matching the async-to-LDS builtin signature:
// (v4i addrspace(1)*, v4i addrspace(3)*, imm offset, imm cpol)
typedef int v4i __attribute__((vector_size(16)));
typedef v4i __attribute__((address_space(1)))* gptr_v4i;
typedef v4i __attribute__((address_space(3)))* lptr_v4i;

// Cholesky (lower) of symmetric 6x6 stored in lower triangle of A, in place.
// Returns product of pre-sqrt pivots (= det). rdiag[j] = 1 / L[j][j].
__device__ __forceinline__ float chol6(float A[6][6], float rdiag[6]) {
  float prod = 1.0f;
#pragma unroll
  for (int j = 0; j < 6; ++j) {
    float x = A[j][j];
#pragma unroll
    for (int k = 0; k < j; ++k) x -= A[j][k] * A[j][k];
    prod *= x;
    float r = RSQF(x);       // 1/sqrt(x)
    rdiag[j] = r;            // == 1 / L[j][j]
    A[j][j] = x * r;         // sqrt(x)
#pragma unroll
    for (int i = j + 1; i < 6; ++i) {
      float s = A[i][j];
#pragma unroll
      for (int k = 0; k < j; ++k) s -= A[i][k] * A[j][k];
      A[i][j] = s * r;
    }
  }
  return prod;
}

__device__ __forceinline__ float wave_reduce_f32(float v) {
#pragma unroll
  for (int off = 16; off > 0; off >>= 1) v += __shfl_down(v, off, 32);
  return v;
}

__global__ __launch_bounds__(TILE) void kl_huber_kernel(
    const float* __restrict__ target, const float* __restrict__ pred,
    float* __restrict__ partials, int nB) {
  __shared__ float sT[TILE * VEC];
  __shared__ float sP[TILE * VEC];
  __shared__ float sRed[8];

  const int tid  = threadIdx.x;
  const int base = blockIdx.x * TILE;           // element index of tile start
  const int n_e  = min(TILE, nB - base);        // elements in this tile
  const int n_f  = n_e * VEC;                   // dwords in tile
  const int n_v4 = n_f >> 2;                    // aligned 16B chunks

  const float* tileT = target + base * VEC;     // tile base: 16B aligned (256*84)
  const float* tileP = pred   + base * VEC;

  // ---- Stage tile into LDS (CDNA5 async global->LDS path) ----
#if __has_builtin(__builtin_amdgcn_global_load_async_to_lds_b128) && \
    __has_builtin(__builtin_amdgcn_s_wait_asynccnt)
  for (int i = tid; i < n_v4; i += TILE) {
    __builtin_amdgcn_global_load_async_to_lds_b128(
        (gptr_v4i)(tileT + 4 * i), (lptr_v4i)(sT + 4 * i), 0, 0);
    __builtin_amdgcn_global_load_async_to_lds_b128(
        (gptr_v4i)(tileP + 4 * i), (lptr_v4i)(sP + 4 * i), 0, 0);
  }
  // generic tail (dwords) — not hit for B=1e6 but keeps kernel correct for any B
  for (int i = (n_v4 << 2) + tid; i < n_f; i += TILE) {
    sT[i] = tileT[i];
    sP[i] = tileP[i];
  }
  __builtin_amdgcn_s_wait_asynccnt(0);
  __syncthreads();
#else
  for (int i = tid; i < n_v4; i += TILE) {
    __builtin_prefetch((const float4*)tileT + i, 0, 1);  // global_prefetch_b8
    ((float4*)sT)[i] = ((const float4*)tileT)[i];
    ((float4*)sP)[i] = ((const float4*)tileP)[i];
  }
  for (int i = (n_v4 << 2) + tid; i < n_f; i += TILE) {
    sT[i] = tileT[i];
    sP[i] = tileP[i];
  }
  __syncthreads();
#endif

  float contrib = 0.0f;
  if (tid < n_e) {
    // LDS reads: stride 21 words; gcd(21,64)=1 -> bank-conflict free
    float tv[VEC], pv[VEC];
#pragma unroll
    for (int k = 0; k < VEC; ++k) {
      tv[k] = sT[tid * VEC + k];
      pv[k] = sP[tid * VEC + k];
    }

    // Huber (delta = 1) over the 21 packed entries
    float hsum = 0.0f;
#pragma unroll
    for (int k = 0; k < VEC; ++k) {
      float d  = pv[k] - tv[k];
      float ad = fabsf(d);
      hsum += (ad < 1.0f) ? 0.5f * d * d : (ad - 0.5f);
    }

    // Unpack upper-tri 21-vector into lower triangle of 6x6 (+ 1e-5 I)
    const int basei[6] = {0, 6, 11, 15, 18, 20};
    float P[6][6], T[6][6];
#pragma unroll
    for (int i = 0; i < 6; ++i) {
#pragma unroll
      for (int j = 0; j <= i; ++j) {
        int k = basei[j] + (i - j);     // packed idx for (row=j, col=i), j<=i
        float e = (i == j) ? 1e-5f : 0.0f;
        P[i][j] = pv[k] + e;
        T[i][j] = tv[k] + e;
      }
    }

    float rT[6], rP[6];
    float prodP = chol6(P, rP);   // P -> Lp (lower), prodP = det(P)
    float prodT = chol6(T, rT);   // T -> Lt (lower), rT[i] = 1/Lt[i][i]

    // M = Lt^{-1} Lp (both lower => M lower); tr(T^{-1}P) = ||M||_F^2
    float tr = 0.0f;
#pragma unroll
    for (int j = 0; j < 6; ++j) {
#pragma unroll
      for (int i = j; i < 6; ++i) {
        float s = P[i][j];
#pragma unroll
        for (int k = j; k < i; ++k) s -= T[i][k] * P[k][j];
        float m = s * rT[i];
        P[i][j] = m;
        tr += m * m;
      }
    }

    // logdetT - logdetP = log(det T / det P) : one transcendental per element
    float kl = 0.5f * (tr - 6.0f + __logf(prodT / prodP));
    contrib  = kl + hsum * (1.0f / 21.0f);
  }

  // ---- deterministic block reduction (wave32 shuffles + LDS) ----
  float v = wave_reduce_f32(contrib);
  const int wid  = tid >> 5;
  const int lane = tid & 31;
  if (lane == 0) sRed[wid] = v;
  __syncthreads();
  if (wid == 0) {
    float r = (lane < (TILE / 32)) ? sRed[lane] : 0.0f;
    r = wave_reduce_f32(r);
    if (lane == 0) partials[blockIdx.x] = r;
  }
}

__global__ __launch_bounds__(256) void final_reduce_kernel(
    const float* __restrict__ partials, int n, float* __restrict__ out,
    double inv_scale) {
  __shared__ double sRed[8];
  double acc = 0.0;
  for (int i = threadIdx.x; i < n; i += 256) acc += (double)partials[i];
#pragma unroll
  for (int off = 16; off > 0; off >>= 1) acc += __shfl_down(acc, off, 32);
  const int wid  = threadIdx.x >> 5;
  const int lane = threadIdx.x & 31;
  if (lane == 0) sRed[wid] = acc;
  __syncthreads();
  if (threadIdx.x == 0) {
    double t = 0.0;
#pragma unroll
    for (int w = 0; w < 8; ++w) t += sRed[w];
    out[0] = (float)(t * inv_scale);
  }
}

extern "C" void kernel_launch(void* const* d_in, const int* in_sizes, int n_in,
                              void* d_out, int out_size, void* d_ws, size_t ws_size,
                              hipStream_t stream) {
  const float* target = (const float*)d_in[0];
  const float* pred   = (const float*)d_in[1];
  const int B = in_sizes[0] / VEC;
  const int nblocks = (B + TILE - 1) / TILE;
  float* partials = (float*)d_ws;

  kl_huber_kernel<<<nblocks, TILE, 0, stream>>>(target, pred, partials, B);
  final_reduce_kernel<<<1, 256, 0, stream>>>(partials, nblocks, (float*)d_out,
                                             1.0 / (double)B);
}